// KERN_21680994910746
// MI455X (gfx1250) — compile-verified
//
#include <hip/hip_runtime.h>
#include <hip/hip_bf16.h>

// Problem sizes (from reference)
#define N_OBJ 512
#define C_OBJ 151
#define C_REL 51
#define N_REL 16384
#define DDIM  4096
#define NMS_THRESH 0.3f

#define OUT_DISTS_OFF 0
#define OUT_PREDS_OFF (N_OBJ * C_OBJ)              // 77312
#define OUT_REL_OFF   (N_OBJ * C_OBJ + N_OBJ)      // 77824

typedef __attribute__((ext_vector_type(2))) float v2f;
typedef __attribute__((ext_vector_type(4))) float v4f;
typedef __attribute__((ext_vector_type(8))) float v8f;

// ---------------------------------------------------------------------------
// Kernel 1: obj_dists2 = obj_logits (passthrough copy)
// ---------------------------------------------------------------------------
__global__ void copy_logits_kernel(const float* __restrict__ src, float* __restrict__ dst) {
    int i = blockIdx.x * blockDim.x + threadIdx.x;
    if (i < N_OBJ * C_OBJ) dst[OUT_DISTS_OFF + i] = src[i];
}

// ---------------------------------------------------------------------------
// Kernel 2: row softmax over 151 classes (one block per object)
// ---------------------------------------------------------------------------
__global__ __launch_bounds__(256) void softmax_kernel(const float* __restrict__ logits,
                                                      float* __restrict__ probs) {
    const int o = blockIdx.x;
    const int t = threadIdx.x;
    __shared__ float red[256];

    float v = (t < C_OBJ) ? logits[o * C_OBJ + t] : -3.0e38f;
    red[t] = v;
    __syncthreads();
    for (int s = 128; s > 0; s >>= 1) {
        if (t < s) red[t] = fmaxf(red[t], red[t + s]);
        __syncthreads();
    }
    float m = red[0];
    __syncthreads();
    float e = (t < C_OBJ) ? expf(v - m) : 0.0f;
    red[t] = e;
    __syncthreads();
    for (int s = 128; s > 0; s >>= 1) {
        if (t < s) red[t] += red[t + s];
        __syncthreads();
    }
    float inv = 1.0f / red[0];
    if (t < C_OBJ) probs[o * C_OBJ + t] = e * inv;
}

// ---------------------------------------------------------------------------
// Kernel 3: per-class NMS. One block (256 threads) per class c.
//   - bitonic sort 512 (score desc, idx asc) in LDS (matches stable argsort(-s))
//   - serial suppression pass, parallel over the 512-wide IoU row
//   - writes masked[o][c] = keep ? probs[o][c] : 0 (class 0 forced 0)
// ---------------------------------------------------------------------------
__global__ __launch_bounds__(256) void nms_kernel(const float* __restrict__ probs,
                                                  const float* __restrict__ boxes,
                                                  float* __restrict__ masked) {
    const int c = blockIdx.x;
    const int t = threadIdx.x;

    __shared__ float ss[N_OBJ];
    __shared__ int   si[N_OBJ];
    __shared__ float bx1[N_OBJ], by1[N_OBJ], bx2[N_OBJ], by2[N_OBJ], ar[N_OBJ];
    __shared__ int   supp[N_OBJ];
    __shared__ int   keep[N_OBJ];

    for (int i = t; i < N_OBJ; i += 256) {
        ss[i] = probs[i * C_OBJ + c];
        si[i] = i;
    }
    __syncthreads();

    // Bitonic sort: descending score, ascending index on ties (stable argsort(-s))
    for (int k = 2; k <= N_OBJ; k <<= 1) {
        for (int j = k >> 1; j > 0; j >>= 1) {
            for (int i = t; i < N_OBJ; i += 256) {
                int ixj = i ^ j;
                if (ixj > i) {
                    bool dirDesc = ((i & k) == 0);
                    float a = ss[i], b = ss[ixj];
                    int ia = si[i], ib = si[ixj];
                    bool a_first = (a > b) || (a == b && ia < ib);
                    bool doswap = dirDesc ? !a_first : a_first;
                    if (doswap) {
                        ss[i] = b; ss[ixj] = a;
                        si[i] = ib; si[ixj] = ia;
                    }
                }
            }
            __syncthreads();
        }
    }

    // Gather boxes in sorted order, precompute areas
    for (int i = t; i < N_OBJ; i += 256) {
        int orig = si[i];
        const float* bp = boxes + ((size_t)orig * C_OBJ + c) * 4;
        float x1 = bp[0], y1 = bp[1], x2 = bp[2], y2 = bp[3];
        bx1[i] = x1; by1[i] = y1; bx2[i] = x2; by2[i] = y2;
        ar[i] = (x2 - x1 + 1.0f) * (y2 - y1 + 1.0f);
        supp[i] = 0;
    }
    __syncthreads();

    // Greedy suppression (matches reference scan over sorted IoU rows)
    for (int i = 0; i < N_OBJ; ++i) {
        int s_i = supp[i];
        __syncthreads();
        if (t == 0) keep[i] = !s_i;
        if (!s_i) {
            float xi1 = bx1[i], yi1 = by1[i], xi2 = bx2[i], yi2 = by2[i], ai = ar[i];
            for (int j = t; j < N_OBJ; j += 256) {
                float xx1 = fmaxf(xi1, bx1[j]);
                float yy1 = fmaxf(yi1, by1[j]);
                float xx2 = fminf(xi2, bx2[j]);
                float yy2 = fminf(yi2, by2[j]);
                float w = fmaxf(xx2 - xx1 + 1.0f, 0.0f);
                float h = fmaxf(yy2 - yy1 + 1.0f, 0.0f);
                float inter = w * h;
                float iou = inter / (ai + ar[j] - inter);
                if (iou > NMS_THRESH) supp[j] = 1;
            }
        }
        __syncthreads();
    }

    // Scatter masked scores back to original index order
    for (int i = t; i < N_OBJ; i += 256) {
        int orig = si[i];
        float v = (keep[i] && c != 0) ? ss[i] : 0.0f;
        masked[orig * C_OBJ + c] = v;
    }
}

// ---------------------------------------------------------------------------
// Kernel 4: obj_preds = argmax(masked[:,1:]) + 1 (first-max, matching jnp)
// ---------------------------------------------------------------------------
__global__ void argmax_kernel(const float* __restrict__ masked, float* __restrict__ out_preds) {
    int o = blockIdx.x * blockDim.x + threadIdx.x;
    if (o >= N_OBJ) return;
    const float* row = masked + (size_t)o * C_OBJ;
    float best = row[1];
    int bi = 1;
    for (int k = 2; k < C_OBJ; ++k) {
        float v = row[k];
        if (v > best) { best = v; bi = k; }
    }
    out_preds[o] = (float)bi;
}

// ---------------------------------------------------------------------------
// Kernel 5: rel_dists = vr @ W^T + b  (M=16384, K=4096, N=51), fp32 WMMA.
// Memory-bound (~260 MB @ 23.3 TB/s ≈ 11 us); V_WMMA_F32_16X16X4_F32 keeps
// fp32 precision and packs 2048 FLOP/instr so the kernel stays BW-limited.
// Block = 128 threads (4 waves). Block tile: 64(M) x 64(K-chunk); wave tile:
// 16(M) x 64(N padded from 51). vr tile staged to LDS with coalesced b128
// loads; W tile staged transposed + K-pair interleaved as float2 so every
// WMMA B operand is a single aligned ds_load_b64 (no scalar loads / movs).
// Next vr K-chunk is prefetched (global_prefetch_b8) during compute.
// ---------------------------------------------------------------------------
__global__ __launch_bounds__(128) void rel_gemm_kernel(const float* __restrict__ vr,
                                                       const float* __restrict__ W,
                                                       const float* __restrict__ bias,
                                                       float* __restrict__ out) {
    constexpr int KC   = 64;   // K chunk
    constexpr int MB   = 64;   // rows per block
    constexpr int LDA  = 68;   // A stride (floats): 16B-aligned rows, bank spread
    constexpr int LDBP = 66;   // B stride (v2f units)

    __shared__ float As[MB * LDA];            // vr tile [row][k]        (17408 B)
    __shared__ v2f   Bsp[(KC / 2) * LDBP];    // W tile  [kpair][n] f2   (16896 B)

    const int t = threadIdx.x;
    const int wave = t >> 5;
    const int lane = t & 31;
    const int seg = lane >> 4;   // 0: lanes 0-15, 1: lanes 16-31
    const int ln  = lane & 15;
    const int m0 = blockIdx.x * MB;

    v8f acc0 = {}, acc1 = {}, acc2 = {}, acc3 = {};

    for (int kb = 0; kb < DDIM; kb += KC) {
        __syncthreads();
        // Stage A: 64 rows x 64 k, 128-bit coalesced loads (8 per thread)
#pragma unroll
        for (int p = 0; p < 8; ++p) {
            int lin = p * 128 + t;           // 0..1023, 16 float4 per row
            int r = lin >> 4, c4 = lin & 15;
            v4f d = *(const v4f*)(vr + (size_t)(m0 + r) * DDIM + kb + c4 * 4);
            *(v4f*)(&As[r * LDA + c4 * 4]) = d;
        }
        // Stage B transposed + pair-interleaved: W[n][kb+4c4..] -> Bsp[kpair][n]
#pragma unroll
        for (int p = 0; p < 7; ++p) {
            int lin = p * 128 + t;           // need lin < 51*16 = 816
            if (lin < C_REL * 16) {
                int n = lin >> 4, c4 = lin & 15;
                v4f d = *(const v4f*)(W + (size_t)n * DDIM + kb + c4 * 4);
                v2f lo = { d.x, d.y };
                v2f hi = { d.z, d.w };
                Bsp[(c4 * 2 + 0) * LDBP + n] = lo;   // K = {4c4,   4c4+1}
                Bsp[(c4 * 2 + 1) * LDBP + n] = hi;   // K = {4c4+2, 4c4+3}
            }
        }
        __syncthreads();

        // Prefetch next vr tile (one 128B line per thread covers 64x64 floats)
        if (kb + KC < DDIM) {
            const float* nxt =
                vr + (size_t)(m0 + (t >> 1)) * DDIM + (kb + KC) + (t & 1) * 32;
            __builtin_prefetch(nxt, 0, 0);
        }

        const float* Ab = &As[(wave * 16 + ln) * LDA];
#pragma unroll
        for (int kk = 0; kk < KC; kk += 4) {
            // A 16x4 f32: lanes 0-15 -> K={kk,kk+1}; lanes 16-31 -> K={kk+2,kk+3}
            v2f a = *(const v2f*)(Ab + kk + 2 * seg);
            // B 4x16 f32 per N-tile: one ds_load_b64 per operand
            const v2f* Bb = &Bsp[((kk >> 1) + seg) * LDBP];
            v2f b0 = Bb[ln];
            v2f b1 = Bb[ln + 16];
            v2f b2 = Bb[ln + 32];
            v2f b3 = Bb[ln + 48];
            acc0 = __builtin_amdgcn_wmma_f32_16x16x4_f32(false, a, false, b0, (short)0, acc0, false, false);
            acc1 = __builtin_amdgcn_wmma_f32_16x16x4_f32(false, a, false, b1, (short)0, acc1, false, false);
            acc2 = __builtin_amdgcn_wmma_f32_16x16x4_f32(false, a, false, b2, (short)0, acc2, false, false);
            acc3 = __builtin_amdgcn_wmma_f32_16x16x4_f32(false, a, false, b3, (short)0, acc3, false, false);
        }
    }

    // Epilogue: C/D layout -> VGPR v: lanes0-15 M=v, lanes16-31 M=v+8
    v8f accs[4] = {acc0, acc1, acc2, acc3};
#pragma unroll
    for (int tn = 0; tn < 4; ++tn) {
        int col = tn * 16 + ln;
        if (col < C_REL) {
            float bv = bias[col];
#pragma unroll
            for (int v = 0; v < 8; ++v) {
                int row = m0 + wave * 16 + v + 8 * seg;
                out[OUT_REL_OFF + (size_t)row * C_REL + col] = accs[tn][v] + bv;
            }
        }
    }
}

// ---------------------------------------------------------------------------
extern "C" void kernel_launch(void* const* d_in, const int* in_sizes, int n_in,
                              void* d_out, int out_size, void* d_ws, size_t ws_size,
                              hipStream_t stream) {
    const float* obj_logits = (const float*)d_in[0];   // 512*151
    const float* vr         = (const float*)d_in[1];   // 16384*4096
    const float* boxes      = (const float*)d_in[2];   // 512*151*4
    const float* W_vr       = (const float*)d_in[3];   // 51*4096
    const float* b_vr       = (const float*)d_in[4];   // 51
    float* out = (float*)d_out;

    float* probs  = (float*)d_ws;                      // 512*151
    float* masked = probs + N_OBJ * C_OBJ;             // 512*151

    copy_logits_kernel<<<(N_OBJ * C_OBJ + 255) / 256, 256, 0, stream>>>(obj_logits, out);
    softmax_kernel<<<N_OBJ, 256, 0, stream>>>(obj_logits, probs);
    nms_kernel<<<C_OBJ, 256, 0, stream>>>(probs, boxes, masked);
    argmax_kernel<<<(N_OBJ + 255) / 256, 256, 0, stream>>>(masked, out + OUT_PREDS_OFF);
    rel_gemm_kernel<<<N_REL / 64, 128, 0, stream>>>(vr, W_vr, b_vr, out);
}